// GINModel_83820581749192
// MI455X (gfx1250) — compile-verified
//
#include <hip/hip_runtime.h>

#define HF 128   // feature / hidden width
#define TOUT 10  // classifier outputs
#define BN_EPS 1e-5f

typedef __attribute__((ext_vector_type(2))) float v2f;
typedef __attribute__((ext_vector_type(8))) float v8f;

__device__ __forceinline__ void atomic_add_f32(float* p, float v) {
  __hip_atomic_fetch_add(p, v, __ATOMIC_RELAXED, __HIP_MEMORY_SCOPE_AGENT);
}

// ---------------- zero a float buffer ----------------
__global__ void zero_kernel(float* __restrict__ p, long long n) {
  long long i = (long long)blockIdx.x * blockDim.x + threadIdx.x;
  if (i < n) p[i] = 0.f;
}

// ------- edge scatter: agg[dst] += x[src]; one wave per edge -------
__global__ void scatter_kernel(const float* __restrict__ X,
                               const int* __restrict__ src,
                               const int* __restrict__ dst,
                               float* __restrict__ AGG, int nE) {
  int wave = blockIdx.x * (blockDim.x >> 5) + (threadIdx.x >> 5);
  int lane = threadIdx.x & 31;
  if (wave >= nE) return;
  int s = src[wave];
  int d = dst[wave];
  const float4 v = *(const float4*)(X + (size_t)s * HF + lane * 4);
  float* out = AGG + (size_t)d * HF + lane * 4;
  atomic_add_f32(out + 0, v.x);
  atomic_add_f32(out + 1, v.y);
  atomic_add_f32(out + 2, v.z);
  atomic_add_f32(out + 3, v.w);
}

// ------- WMMA fp32 GEMM: Y = prologue(X) @ W + bias (128x128 weights) -------
// MODE 0: prologue = X + AGG                    (GIN conv input; no epilogue act)
// MODE 1: prologue = relu(X*scale + shift)      (BN+ReLU); epilogue = relu
// Block: 256 threads = 8 waves; each block does 16 rows x 128 cols,
// each wave owns one 16x16 column tile. Safe for in-place X == Y.
template <int MODE>
__global__ void gemm_kernel(const float* __restrict__ X,
                            const float* __restrict__ AGG,
                            const float* __restrict__ W,     // [128(in) x 128(out)] row-major
                            const float* __restrict__ bias,  // [128]
                            const float* __restrict__ scale, // [128] (MODE 1)
                            const float* __restrict__ shift, // [128] (MODE 1)
                            float* __restrict__ Y, int n_rows) {
  __shared__ float s[16][HF + 4];  // stride 132 floats -> conflict-free column reads
  const int tid = threadIdx.x;
  const int r0 = blockIdx.x * 16;

  // stage 16x128 activation tile in LDS with fused prologue
  for (int i = tid; i < 16 * HF; i += 256) {
    int row = i >> 7, col = i & (HF - 1);
    int gr = r0 + row;
    float v = 0.f;
    if (gr < n_rows) {
      v = X[(size_t)gr * HF + col];
      if (MODE == 0) v += AGG[(size_t)gr * HF + col];
      else           v = fmaxf(fmaf(v, scale[col], shift[col]), 0.f);
    }
    s[row][col] = v;
  }
  __syncthreads();

  const int wave = tid >> 5;
  const int lane = tid & 31;
  const int half = lane >> 4;   // 0: lanes 0-15, 1: lanes 16-31
  const int lrow = lane & 15;   // M index for A frag / N index for B,C frags
  const int col  = wave * 16 + lrow;

  v8f acc = {};
  for (int k0 = 0; k0 < HF; k0 += 4) {
    const int ka = k0 + half * 2;          // ISA A/B frag layout for 16x16x4 f32
    v2f a, b;
    a.x = s[lrow][ka];
    a.y = s[lrow][ka + 1];
    b.x = W[(size_t)(ka)     * HF + col];
    b.y = W[(size_t)(ka + 1) * HF + col];
    acc = __builtin_amdgcn_wmma_f32_16x16x4_f32(false, a, false, b, (short)0, acc,
                                                false, false);
  }

  const float bcol = bias[col];
#pragma unroll
  for (int v = 0; v < 8; ++v) {
    int gr = r0 + half * 8 + v;            // C/D layout: VGPR v -> row v + 8*half
    if (gr < n_rows) {
      float val = acc[v] + bcol;
      if (MODE == 1) val = fmaxf(val, 0.f);
      Y[(size_t)gr * HF + col] = val;
    }
  }
}

// ------- per-feature sum / sum-of-squares for BatchNorm -------
__global__ void stats_kernel(const float* __restrict__ H, float* __restrict__ sum,
                             float* __restrict__ sumsq, int n_rows) {
  const int col = threadIdx.x;             // blockDim = 128
  const int r0 = blockIdx.x * 256;
  int r1 = r0 + 256; if (r1 > n_rows) r1 = n_rows;
  float a = 0.f, b = 0.f;
  for (int r = r0; r < r1; ++r) {
    float v = H[(size_t)r * HF + col];
    a += v;
    b = fmaf(v, v, b);
  }
  atomic_add_f32(sum + col, a);
  atomic_add_f32(sumsq + col, b);
}

__global__ void finalize_bn_kernel(const float* __restrict__ sum,
                                   const float* __restrict__ sumsq,
                                   const float* __restrict__ gamma,
                                   const float* __restrict__ beta,
                                   float* __restrict__ scale,
                                   float* __restrict__ shift, int n_rows) {
  int f = threadIdx.x;                     // 128 threads
  float inv_n = 1.f / (float)n_rows;
  float mean = sum[f] * inv_n;
  float var = sumsq[f] * inv_n - mean * mean;  // biased variance (training BN)
  float sc = gamma[f] * rsqrtf(var + BN_EPS);
  scale[f] = sc;
  shift[f] = fmaf(-mean, sc, beta[f]);
}

// ------- graph pooling: pooled[batch[n]] += h[n]; one wave per node -------
__global__ void pool_kernel(const float* __restrict__ H, const int* __restrict__ batch,
                            float* __restrict__ pooled, int n_rows) {
  int wave = blockIdx.x * (blockDim.x >> 5) + (threadIdx.x >> 5);
  int lane = threadIdx.x & 31;
  if (wave >= n_rows) return;
  int g = batch[wave];
  const float4 v = *(const float4*)(H + (size_t)wave * HF + lane * 4);
  float* out = pooled + (size_t)g * HF + lane * 4;
  atomic_add_f32(out + 0, v.x);
  atomic_add_f32(out + 1, v.y);
  atomic_add_f32(out + 2, v.z);
  atomic_add_f32(out + 3, v.w);
}

// ------- final classifier: out[g] = pooled[g] @ Wl + bl  (128 -> 10) -------
__global__ void final_kernel(const float* __restrict__ pooled,
                             const float* __restrict__ Wl,
                             const float* __restrict__ bl, float* __restrict__ out) {
  __shared__ float row[HF];
  int g = blockIdx.x;                      // blockDim = 128
  row[threadIdx.x] = pooled[(size_t)g * HF + threadIdx.x];
  __syncthreads();
  int t = threadIdx.x;
  if (t < TOUT) {
    float acc = bl[t];
    for (int k = 0; k < HF; ++k) acc = fmaf(row[k], Wl[k * TOUT + t], acc);
    out[g * TOUT + t] = acc;
  }
}

extern "C" void kernel_launch(void* const* d_in, const int* in_sizes, int n_in,
                              void* d_out, int out_size, void* d_ws, size_t ws_size,
                              hipStream_t stream) {
  const float* x   = (const float*)d_in[0];
  const int*   ei  = (const int*)d_in[1];
  const int* batch = (const int*)d_in[2];
  const float* W1a = (const float*)d_in[3];
  const float* b1a = (const float*)d_in[4];
  const float* g1  = (const float*)d_in[5];
  const float* bt1 = (const float*)d_in[6];
  const float* W1b = (const float*)d_in[7];
  const float* b1b = (const float*)d_in[8];
  const float* W2a = (const float*)d_in[9];
  const float* b2a = (const float*)d_in[10];
  const float* g2  = (const float*)d_in[11];
  const float* bt2 = (const float*)d_in[12];
  const float* W2b = (const float*)d_in[13];
  const float* b2b = (const float*)d_in[14];
  const float* Wl  = (const float*)d_in[15];
  const float* bl  = (const float*)d_in[16];

  const int nN = in_sizes[0] / HF;   // 100000
  const int nE = in_sizes[1] / 2;    // 1600000
  const int nG = out_size / TOUT;    // 512
  const int* src = ei;
  const int* dst = ei + nE;

  // workspace layout
  float* agg    = (float*)d_ws;                     // N*128
  float* h1     = agg + (size_t)nN * HF;            // N*128 (reused across layers)
  float* sum    = h1 + (size_t)nN * HF;             // 128
  float* sumsq  = sum + HF;                         // 128
  float* scale  = sumsq + HF;                       // 128
  float* shift  = scale + HF;                       // 128
  float* pooled = shift + HF;                       // G*128

  const long long aggN = (long long)nN * HF;
  const unsigned zeroBlocks  = (unsigned)((aggN + 255) / 256);
  const unsigned edgeBlocks  = (unsigned)((nE + 7) / 8);   // 8 waves/block, 1 edge/wave
  const unsigned rowTiles    = (unsigned)((nN + 15) / 16);
  const unsigned nodeBlocks  = (unsigned)((nN + 7) / 8);
  const unsigned statsBlocks = (unsigned)((nN + 255) / 256);

  // ---------------- Layer 1 ----------------
  zero_kernel<<<zeroBlocks, 256, 0, stream>>>(agg, aggN);
  zero_kernel<<<1, 256, 0, stream>>>(sum, 256);  // sum+sumsq contiguous
  scatter_kernel<<<edgeBlocks, 256, 0, stream>>>(x, src, dst, agg, nE);
  gemm_kernel<0><<<rowTiles, 256, 0, stream>>>(x, agg, W1a, b1a, nullptr, nullptr, h1, nN);
  stats_kernel<<<statsBlocks, 128, 0, stream>>>(h1, sum, sumsq, nN);
  finalize_bn_kernel<<<1, 128, 0, stream>>>(sum, sumsq, g1, bt1, scale, shift, nN);
  gemm_kernel<1><<<rowTiles, 256, 0, stream>>>(h1, nullptr, W1b, b1b, scale, shift, h1, nN);

  // ---------------- Layer 2 ----------------
  zero_kernel<<<zeroBlocks, 256, 0, stream>>>(agg, aggN);
  zero_kernel<<<1, 256, 0, stream>>>(sum, 256);
  scatter_kernel<<<edgeBlocks, 256, 0, stream>>>(h1, src, dst, agg, nE);
  gemm_kernel<0><<<rowTiles, 256, 0, stream>>>(h1, agg, W2a, b2a, nullptr, nullptr, h1, nN);
  stats_kernel<<<statsBlocks, 128, 0, stream>>>(h1, sum, sumsq, nN);
  finalize_bn_kernel<<<1, 128, 0, stream>>>(sum, sumsq, g2, bt2, scale, shift, nN);
  gemm_kernel<1><<<rowTiles, 256, 0, stream>>>(h1, nullptr, W2b, b2b, scale, shift, h1, nN);

  // ---------------- Pool + classify ----------------
  zero_kernel<<<(unsigned)(((long long)nG * HF + 255) / 256), 256, 0, stream>>>(
      pooled, (long long)nG * HF);
  pool_kernel<<<nodeBlocks, 256, 0, stream>>>(h1, batch, pooled, nN);
  final_kernel<<<(unsigned)nG, 128, 0, stream>>>(pooled, Wl, bl, (float*)d_out);
}